// VoVNetBEVTransformer_23691039604926
// MI455X (gfx1250) — compile-verified
//
#include <hip/hip_runtime.h>

// ---------------- problem constants (from reference) ----------------
#define B_   4
#define N_   6
#define D_   41
#define FH_  8
#define FW_  22
#define C_   128
#define NX_  200
#define NY_  200
#define NPTS (B_ * N_ * D_ * FH_ * FW_)      // 173184
#define S_   (B_ * NX_ * NY_)                // 160000 voxels (NZ==1)
#define K_   16                              // per-voxel list capacity
#define TILES_ ((NY_ + 31) / 32)             // 7 gy-tiles per (b,gx)
#define GATHER_BLOCKS (B_ * NX_ * TILES_)    // 5600
#define TPAD_ 132                            // LDS tile row stride (floats)

#define OUT_FLOATS (B_ * C_ * NX_ * NY_)     // 20,480,000
#define ZERO_BLOCKS (OUT_FLOATS / (256 * 4)) // 20000 (fallback path)
#define SCATTER_BLOCKS (NPTS / 16)           // 10824 (fallback path)

// ---------------- CDNA5 async-to-LDS feature detection ----------------
#if defined(__HIP_DEVICE_COMPILE__)
#define HAS_ASYNC_LDS  __has_builtin(__builtin_amdgcn_global_load_async_to_lds_b128)
#define HAS_WAIT_ASYNC __has_builtin(__builtin_amdgcn_s_wait_asynccnt)
#else
#define HAS_ASYNC_LDS 0
#define HAS_WAIT_ASYNC 0
#endif

typedef int v4i __attribute__((ext_vector_type(4)));
typedef __attribute__((address_space(1))) v4i* gp_v4i;   // global (AS1) v4i*
typedef __attribute__((address_space(3))) v4i* lp_v4i;   // LDS (AS3) v4i*

#if HAS_ASYNC_LDS
#define ASYNC_STAGE(src, dst)                                        \
    __builtin_amdgcn_global_load_async_to_lds_b128(                  \
        (gp_v4i)(unsigned long long)(const void*)(src),              \
        (lp_v4i)(unsigned int)(unsigned long long)(void*)(dst), 0, 0)
#if HAS_WAIT_ASYNC
#define WAIT_ASYNC(n) __builtin_amdgcn_s_wait_asynccnt(n)
#else
#define WAIT_ASYNC(n) asm volatile("s_wait_asynccnt " #n ::: "memory")
#endif
#endif

// ---------------- 3x3 inverse (adjugate / det) ----------------
__device__ __forceinline__ void inv3(const float* __restrict__ m, float* __restrict__ o) {
    float c00 = m[4] * m[8] - m[5] * m[7];
    float c01 = m[5] * m[6] - m[3] * m[8];
    float c02 = m[3] * m[7] - m[4] * m[6];
    float id  = 1.0f / (m[0] * c00 + m[1] * c01 + m[2] * c02);
    o[0] = c00 * id;
    o[1] = (m[2] * m[7] - m[1] * m[8]) * id;
    o[2] = (m[1] * m[5] - m[2] * m[4]) * id;
    o[3] = c01 * id;
    o[4] = (m[0] * m[8] - m[2] * m[6]) * id;
    o[5] = (m[2] * m[3] - m[0] * m[5]) * id;
    o[6] = c02 * id;
    o[7] = (m[1] * m[6] - m[0] * m[7]) * id;
    o[8] = (m[0] * m[4] - m[1] * m[3]) * id;
}

// ---------------- per-point frustum -> voxel geometry ----------------
// P layout per (b*N+n), stride 24 floats:
//   [0..8] Minv=inv(post_rots)  [9..17] comb=rots@inv(intrins)
//   [18..20] post_trans         [21..23] trans
struct Geo { int b, gx, gy; bool kept; };

__device__ __forceinline__ Geo point_geom(int p, const float* __restrict__ P) {
    int t = p;
    const int w = t % FW_; t /= FW_;
    const int h = t % FH_; t /= FH_;
    const int d = t % D_;  t /= D_;          // t = b*N_ + n
    const float* pp = P + t * 24;

    const float u   = (float)w * (351.0f / 21.0f);   // linspace(0,351,22)
    const float v   = (float)h * (127.0f / 7.0f);    // linspace(0,127,8)
    const float dep = 4.0f + (float)d;               // arange(4,45,1)

    const float tx = u - pp[18], ty = v - pp[19], tz = dep - pp[20];
    float qx = pp[0] * tx + pp[1] * ty + pp[2] * tz;
    float qy = pp[3] * tx + pp[4] * ty + pp[5] * tz;
    const float qz = pp[6] * tx + pp[7] * ty + pp[8] * tz;
    qx *= qz; qy *= qz;
    const float ex = pp[9]  * qx + pp[10] * qy + pp[11] * qz + pp[21];
    const float ey = pp[12] * qx + pp[13] * qy + pp[14] * qz + pp[22];
    const float ez = pp[15] * qx + pp[16] * qy + pp[17] * qz + pp[23];

    Geo g;
    g.gx = (int)((ex + 50.0f) / 0.5f);    // trunc-toward-zero == astype(int32)
    g.gy = (int)((ey + 50.0f) / 0.5f);
    const int gz = (int)((ez + 10.0f) / 20.0f);
    g.b  = t / N_;
    g.kept = (g.gx >= 0) && (g.gx < NX_) && (g.gy >= 0) && (g.gy < NY_) && (gz == 0);
    return g;
}

// ---------------- kernel: per-(b,n) camera matrices ----------------
__global__ void setup_params(const float* __restrict__ rots,
                             const float* __restrict__ trans,
                             const float* __restrict__ intrins,
                             const float* __restrict__ post_rots,
                             const float* __restrict__ post_trans,
                             float* __restrict__ P) {
    int id = threadIdx.x;
    if (id >= B_ * N_) return;
    float Minv[9], Kinv[9];
    inv3(post_rots + id * 9, Minv);
    inv3(intrins + id * 9, Kinv);
    const float* R = rots + id * 9;
    float* p = P + id * 24;
    for (int i = 0; i < 9; ++i) p[i] = Minv[i];
    for (int i = 0; i < 3; ++i)
        for (int j = 0; j < 3; ++j) {
            float s = 0.0f;
            for (int k = 0; k < 3; ++k) s += R[i * 3 + k] * Kinv[k * 3 + j];
            p[9 + i * 3 + j] = s;
        }
    for (int i = 0; i < 3; ++i) p[18 + i] = post_trans[id * 3 + i];
    for (int i = 0; i < 3; ++i) p[21 + i] = trans[id * 3 + i];
}

// ---------------- binned-gather path ----------------
__global__ __launch_bounds__(256) void zero_bins(int* __restrict__ a, int n) {
    int i = blockIdx.x * 256 + threadIdx.x;
    if (i < n) a[i] = 0;
}

__global__ __launch_bounds__(256) void bin_points(const float* __restrict__ P,
                                                  int* __restrict__ cnt,
                                                  int* __restrict__ list,
                                                  int* __restrict__ ovf,
                                                  int* __restrict__ cursor) {
    const int p = blockIdx.x * 256 + threadIdx.x;
    if (p >= NPTS) return;
    const Geo g = point_geom(p, P);
    if (!g.kept) return;
    const int v = (g.b * NX_ + g.gx) * NY_ + g.gy;
    const int slot = atomicAdd(&cnt[v], 1);
    if (slot < K_) {
        list[(size_t)v * K_ + slot] = p;
    } else {
        const int pos = atomicAdd(cursor, 1);
        ovf[pos] = p;
    }
}

// One block per (b, gx, 32-wide gy tile). Wave wv accumulates voxels
// gy = gy0 + wv*4 .. +3 (coalesced 512B feature reads, async double-buffered
// through LDS), then the block writes the 32x128 tile transposed & coalesced.
__global__ __launch_bounds__(256) void gather_write(const float* __restrict__ feats,
                                                    const int* __restrict__ cnt,
                                                    const int* __restrict__ list,
                                                    float* __restrict__ out) {
    __shared__ float tile[32 * TPAD_];          // 16,896 B
#if HAS_ASYNC_LDS
    __shared__ float stage[8 * 2 * C_];         // 8 waves x 2 slots x 512 B
#endif
    const int lane = threadIdx.x & 31;
    const int wv   = threadIdx.x >> 5;

    int blk = blockIdx.x;
    const int tile_i = blk % TILES_; blk /= TILES_;
    const int gx = blk % NX_;
    const int b  = blk / NX_;
    const int gy0 = tile_i * 32;

    #pragma unroll
    for (int i4 = 0; i4 < 4; ++i4) {
        const int ty = wv * 4 + i4;
        const int gy = gy0 + ty;
        if (gy < NY_) {
            const int v = (b * NX_ + gx) * NY_ + gy;
            int n = cnt[v]; n = (n < K_) ? n : K_;
            const int* lst = list + (size_t)v * K_;
            float4 acc = make_float4(0.0f, 0.0f, 0.0f, 0.0f);
#if HAS_ASYNC_LDS
            float* slot0 = &stage[(wv * 2 + 0) * C_];
            float* slot1 = &stage[(wv * 2 + 1) * C_];
            if (n > 0) ASYNC_STAGE(feats + (size_t)lst[0] * C_ + lane * 4, slot0 + lane * 4);
            for (int i = 0; i < n; ++i) {
                float* cur = (i & 1) ? slot1 : slot0;
                if (i + 1 < n) {
                    float* nxt = ((i + 1) & 1) ? slot1 : slot0;
                    ASYNC_STAGE(feats + (size_t)lst[i + 1] * C_ + lane * 4, nxt + lane * 4);
                    WAIT_ASYNC(1);
                } else {
                    WAIT_ASYNC(0);
                }
                const float4 f = *(const float4*)(cur + lane * 4);
                acc.x += f.x; acc.y += f.y; acc.z += f.z; acc.w += f.w;
            }
#else
            for (int i = 0; i < n; ++i) {
                const float4 f = *(const float4*)(feats + (size_t)lst[i] * C_ + lane * 4);
                acc.x += f.x; acc.y += f.y; acc.z += f.z; acc.w += f.w;
            }
#endif
            *(float4*)&tile[ty * TPAD_ + lane * 4] = acc;   // 16B-aligned (528*ty+16*lane)
        }
    }
    __syncthreads();

    // transpose write: e = c*32 + ty; consecutive threads -> consecutive gy (128B lines)
    #pragma unroll
    for (int j = 0; j < 16; ++j) {
        const int e  = j * 256 + threadIdx.x;
        const int c  = e >> 5;
        const int ty = e & 31;
        const int gy = gy0 + ty;
        if (gy < NY_)
            out[((size_t)(b * C_ + c) * NX_ + gx) * NY_ + gy] = tile[ty * TPAD_ + c];
    }
}

// Rare >K points: fp atomics on top of the plain-stored output.
__global__ __launch_bounds__(256) void overflow_scatter(const float* __restrict__ feats,
                                                        const float* __restrict__ P,
                                                        const int* __restrict__ ovf,
                                                        const int* __restrict__ cursor,
                                                        float* __restrict__ out) {
    const int lane = threadIdx.x & 31;
    const int wv   = threadIdx.x >> 5;
    const int nwaves = gridDim.x * 8;
    const int count = cursor[0];
    for (int wid = blockIdx.x * 8 + wv; wid < count; wid += nwaves) {
        const int p = ovf[wid];
        const Geo g = point_geom(p, P);   // kept by construction
        const float4 f = *(const float4*)(feats + (size_t)p * C_ + lane * 4);
        float* o = out + ((size_t)(g.b * C_ + lane * 4) * NX_ + g.gx) * NY_ + g.gy;
        const int stride = NX_ * NY_;
        unsafeAtomicAdd(o,              f.x);
        unsafeAtomicAdd(o + stride,     f.y);
        unsafeAtomicAdd(o + 2 * stride, f.z);
        unsafeAtomicAdd(o + 3 * stride, f.w);
    }
}

// ---------------- fallback path (small workspace): direct atomic scatter ----
__global__ __launch_bounds__(256) void zero_out(float4* __restrict__ out) {
    int idx = blockIdx.x * 256 + threadIdx.x;
    out[idx] = make_float4(0.0f, 0.0f, 0.0f, 0.0f);
}

__global__ __launch_bounds__(256) void bev_scatter(const float* __restrict__ feats,
                                                   const float* __restrict__ P,
                                                   float* __restrict__ out) {
    __shared__ float lds[8 * 2 * C_];
    const int lane = threadIdx.x & 31;
    const int wv   = threadIdx.x >> 5;
    const int pbase = (blockIdx.x * 8 + wv) * 2;
    float* stage = &lds[wv * (2 * C_)];

    #pragma unroll
    for (int k = 0; k < 2; ++k) {
        const float* src = feats + (size_t)(pbase + k) * C_ + lane * 4;
        float* dst = stage + k * C_ + lane * 4;
#if HAS_ASYNC_LDS
        ASYNC_STAGE(src, dst);
#else
        *(float4*)dst = *(const float4*)src;
#endif
    }

    auto process = [&](int k) __attribute__((always_inline)) {
        const int p = pbase + k;
        const Geo g = point_geom(p, P);
        const float4 f = *(const float4*)(stage + k * C_ + lane * 4);
        if (g.kept) {
            float* o = out + ((size_t)(g.b * C_ + lane * 4) * NX_ + g.gx) * NY_ + g.gy;
            const int stride = NX_ * NY_;
            unsafeAtomicAdd(o,              f.x);
            unsafeAtomicAdd(o + stride,     f.y);
            unsafeAtomicAdd(o + 2 * stride, f.z);
            unsafeAtomicAdd(o + 3 * stride, f.w);
        }
    };

#if HAS_ASYNC_LDS
    WAIT_ASYNC(1);
#endif
    process(0);
#if HAS_ASYNC_LDS
    WAIT_ASYNC(0);
#endif
    process(1);
}

// ---------------- launcher ----------------
extern "C" void kernel_launch(void* const* d_in, const int* in_sizes, int n_in,
                              void* d_out, int out_size, void* d_ws, size_t ws_size,
                              hipStream_t stream) {
    const float* cam_feats  = (const float*)d_in[0];
    const float* rots       = (const float*)d_in[1];
    const float* trans      = (const float*)d_in[2];
    const float* intrins    = (const float*)d_in[3];
    const float* post_rots  = (const float*)d_in[4];
    const float* post_trans = (const float*)d_in[5];
    float* out = (float*)d_out;

    char* ws = (char*)d_ws;
    float* params = (float*)ws;                                   // 2,304 B (pad to 4096)
    int*   cnt    = (int*)(ws + 4096);                            // 640,000 B
    int*   cursor = (int*)(ws + 4096 + 640000);                   // 4 B (contiguous after cnt)
    int*   list   = (int*)(ws + 4096 + 640000 + 64);              // 10,240,000 B
    int*   ovf    = (int*)(ws + 4096 + 640000 + 64 + 10240000);   // 692,736 B
    const size_t need = 4096 + 640000 + 64 + 10240000 + (size_t)NPTS * 4;

    setup_params<<<1, 32, 0, stream>>>(rots, trans, intrins, post_rots, post_trans, params);

    if (ws_size >= need) {
        // binned gather path: ~173K int atomics + streaming, coalesced writes
        zero_bins<<<(S_ + 1 + 255) / 256, 256, 0, stream>>>(cnt, S_ + 1);
        bin_points<<<(NPTS + 255) / 256, 256, 0, stream>>>(params, cnt, list, ovf, cursor);
        gather_write<<<GATHER_BLOCKS, 256, 0, stream>>>(cam_feats, cnt, list, out);
        overflow_scatter<<<1024, 256, 0, stream>>>(cam_feats, params, ovf, cursor, out);
    } else {
        // fallback: direct fp-atomic scatter
        zero_out<<<ZERO_BLOCKS, 256, 0, stream>>>((float4*)out);
        bev_scatter<<<SCATTER_BLOCKS, 256, 0, stream>>>(cam_feats, params, out);
    }
}